// AmplitudeEncodingClassifier_42073499631639
// MI455X (gfx1250) — compile-verified
//
#include <hip/hip_runtime.h>
#include <hip/hip_bf16.h>
#include <math.h>

typedef __attribute__((ext_vector_type(2))) float v2f;
typedef __attribute__((ext_vector_type(8))) float v8f;

#define B_TOT   4096
#define L_IN    4448
#define L1      1112          // conv1 out length
#define LP      278           // after maxpool4
#define L2LEN   139           // conv2 out length
#define SD      256
#define XPAD_N  4624          // padded flux (max read = 16*287 + 27 = 4619)
#define PSTR    296           // pooled row stride (max read = 2*143+6 = 292)

// ---------------------------------------------------------------------------
// Kernel 1: fused conv1 + BN + ReLU + MaxPool4 + conv2 + BN + ReLU + adaptive
// avg-pool(8).  One block (256 threads = 8 waves) per batch sample.
// Both convs run as implicit-im2col GEMMs on fp32 WMMA 16x16x4.
// conv1 tiles N over POOLED positions with 4 offset-accumulators, so MaxPool4
// is 3 lane-local v_max ops (no shuffles / bpermute).
// ---------------------------------------------------------------------------
__global__ __launch_bounds__(256) void extract_kernel(
    const float* __restrict__ flux,
    const float* __restrict__ w1g, const float* __restrict__ g1, const float* __restrict__ bb1,
    const float* __restrict__ w2g, const float* __restrict__ g2, const float* __restrict__ bb2,
    float* __restrict__ feat)
{
    __shared__ float smem[13280];
    float* xpad   = smem;                 // 4624 (later reused as out2: 32*139=4448)
    float* pooled = smem + XPAD_N;        // 16 * 296 = 4736 (pad 3 left, zeros right)
    float* w2s    = pooled + 16 * PSTR;   // 3584 = 32*112, K-major (ic*7+kk)
    float* w1s    = w2s + 3584;           // 240
    float* s1     = w1s + 240;            // 16
    float* t1     = s1 + 16;              // 16
    float* s2     = t1 + 16;              // 32
    float* t2     = s2 + 32;              // 32

    const int tid  = threadIdx.x;
    const int lane = tid & 31, wave = tid >> 5;
    const int l15  = lane & 15, half = lane >> 4;
    const int b    = blockIdx.x;
    const float bninv = rsqrtf(1.f + 1e-5f);

    for (int i = tid; i < XPAD_N;    i += 256) xpad[i]   = 0.f;
    for (int i = tid; i < 16 * PSTR; i += 256) pooled[i] = 0.f;
    for (int i = tid; i < 240;       i += 256) w1s[i] = w1g[i];
    for (int i = tid; i < 3584;      i += 256) w2s[i] = w2g[i];
    if (tid < 16)      { s1[tid] = g1[tid] * bninv; t1[tid] = bb1[tid]; }
    else if (tid < 48) { int c = tid - 16; s2[c] = g2[c] * bninv; t2[c] = bb2[c]; }
    __syncthreads();
    for (int i = tid; i < L_IN; i += 256) xpad[7 + i] = flux[b * L_IN + i];
    __syncthreads();

    // ---- conv1 as WMMA GEMM, N = pooled positions (18 tiles of 16).
    // acc_oo(k-sum) over B_oo(k,p) = xpad[16p + 4*oo + k]; pool = max of 4 accs.
    {
        v2f afr[4];                       // loop-invariant weight fragments
        #pragma unroll
        for (int s = 0; s < 4; ++s) {
            int k = s * 4 + half * 2;
            afr[s][0] = (k     < 15) ? w1s[l15 * 15 + k]     : 0.f;
            afr[s][1] = (k + 1 < 15) ? w1s[l15 * 15 + k + 1] : 0.f;
        }
        float sc[8], tb[8];
        #pragma unroll
        for (int r = 0; r < 8; ++r) { sc[r] = s1[r + 8*half]; tb[r] = t1[r + 8*half]; }

        for (int t = wave; t < 18; t += 8) {
            const int p0   = t * 16;
            const int base = 16 * (p0 + l15) + half * 2;
            v2f bw[7];                    // shared B pairs: offsets 4u, u=0..6
            #pragma unroll
            for (int u = 0; u < 7; ++u) {
                bw[u][0] = xpad[base + 4*u];
                bw[u][1] = xpad[base + 4*u + 1];
            }
            v8f a0 = {}, a1 = {}, a2 = {}, a3 = {};
            #pragma unroll
            for (int s = 0; s < 4; ++s) {
                a0 = __builtin_amdgcn_wmma_f32_16x16x4_f32(false, afr[s], false, bw[s],   (short)0, a0, false, false);
                a1 = __builtin_amdgcn_wmma_f32_16x16x4_f32(false, afr[s], false, bw[s+1], (short)0, a1, false, false);
                a2 = __builtin_amdgcn_wmma_f32_16x16x4_f32(false, afr[s], false, bw[s+2], (short)0, a2, false, false);
                a3 = __builtin_amdgcn_wmma_f32_16x16x4_f32(false, afr[s], false, bw[s+3], (short)0, a3, false, false);
            }
            const int  p  = p0 + l15;
            const bool ok = p < LP;
            #pragma unroll
            for (int r = 0; r < 8; ++r) {
                int row = r + 8 * half;
                float v0 = fmaxf(fmaf(a0[r], sc[r], tb[r]), 0.f);
                float v1 = fmaxf(fmaf(a1[r], sc[r], tb[r]), 0.f);
                float v2 = fmaxf(fmaf(a2[r], sc[r], tb[r]), 0.f);
                float v3 = fmaxf(fmaf(a3[r], sc[r], tb[r]), 0.f);
                float m  = fmaxf(fmaxf(v0, v1), fmaxf(v2, v3));
                if (ok) pooled[row * PSTR + 3 + p] = m;
            }
        }
    }
    __syncthreads();

    // ---- conv2 as WMMA GEMM: M=32ch (2 tiles), K=112 (28 steps), N=139 (9 tiles)
    // B(k,n) = pooled[k/7][2n + k%7]; A = w2s (already K-major). out2 aliases xpad.
    float* out2 = xpad;
    {
        for (int t = wave; t < 18; t += 8) {
            const int mt = t & 1, nt = t >> 1;
            const int c0 = mt * 16, o0 = nt * 16;
            const int n2 = 2 * (o0 + l15);
            const float* wrow = &w2s[(c0 + l15) * 112];
            v8f acc = {};
            #pragma unroll
            for (int s = 0; s < 28; ++s) {
                int ka  = s * 4 + half * 2;
                int ka1 = ka + 1;
                int ic0 = ka / 7,  kk0 = ka - ic0 * 7;
                int ic1 = ka1 / 7, kk1 = ka1 - ic1 * 7;
                v2f a, bf;
                a[0]  = wrow[ka];
                a[1]  = wrow[ka1];
                bf[0] = pooled[ic0 * PSTR + n2 + kk0];
                bf[1] = pooled[ic1 * PSTR + n2 + kk1];
                acc = __builtin_amdgcn_wmma_f32_16x16x4_f32(
                        false, a, false, bf, (short)0, acc, false, false);
            }
            #pragma unroll
            for (int r = 0; r < 8; ++r) {
                int row = c0 + r + 8 * half;
                float v = fmaf(acc[r], s2[row], t2[row]);
                v = v > 0.f ? v : 0.f;
                if (o0 + l15 < L2LEN)
                    out2[row * L2LEN + o0 + l15] = v;
            }
        }
    }
    __syncthreads();

    // ---- adaptive avg pool to 8, features ordered c*8 + i
    {
        int c = tid >> 3, i = tid & 7;
        int st = (i * L2LEN) >> 3;
        int en = ((i + 1) * L2LEN + 7) >> 3;
        float acc = 0.f;
        for (int j = st; j < en; ++j) acc += out2[c * L2LEN + j];
        feat[b * SD + c * 8 + i] = acc / (float)(en - st);
    }
}

// ---------------------------------------------------------------------------
// Kernel 2: proj1 (256->64, ReLU) + proj2 (64->256) with fp32 WMMA 16x16x4,
// fused with per-row L2 normalization (+ uniform fallback).
// One block = 16 batch rows, 4 waves.
// ---------------------------------------------------------------------------
__global__ __launch_bounds__(128) void proj_norm_kernel(
    const float* __restrict__ feat,
    const float* __restrict__ w1, const float* __restrict__ b1,
    const float* __restrict__ w2, const float* __restrict__ b2,
    float* __restrict__ xout)
{
    __shared__ float sf[16][260];   // feature tile (padded: stride%64==4)
    __shared__ float sh[16][68];    // hidden
    __shared__ float sx[16][260];   // output tile
    __shared__ float srow[16][8];
    __shared__ float sscale[16];
    __shared__ int   suni[16];

    const int tid  = threadIdx.x;
    const int lane = tid & 31, wave = tid >> 5;
    const int m0   = blockIdx.x * 16;
    const int half = lane >> 4;
    const int l15  = lane & 15;

    for (int idx = tid; idx < 16 * 64; idx += 128) {
        int r = idx >> 6, c4 = idx & 63;
        ((float4*)&sf[r][0])[c4] = ((const float4*)(feat + (m0 + r) * SD))[c4];
    }
    __syncthreads();

    // ---- GEMM1: sh = relu(sf[16,256] @ w1[64,256]^T + b1)
    {
        const int n0 = wave * 16;
        v8f acc = {};
        for (int k0 = 0; k0 < 256; k0 += 4) {
            int ka = k0 + half * 2;
            v2f a, bf;
            a[0] = sf[l15][ka];  a[1] = sf[l15][ka + 1];
            const float* wp = w1 + (n0 + l15) * 256 + ka;
            bf[0] = wp[0];       bf[1] = wp[1];
            acc = __builtin_amdgcn_wmma_f32_16x16x4_f32(
                    false, a, false, bf, (short)0, acc, false, false);
        }
        float bias = b1[n0 + l15];
        #pragma unroll
        for (int r = 0; r < 8; ++r) {
            int row = r + 8 * half;
            float v = acc[r] + bias;
            sh[row][n0 + l15] = v > 0.f ? v : 0.f;
        }
    }
    __syncthreads();

    // ---- GEMM2: sx = sh[16,64] @ w2[256,64]^T + b2
    #pragma unroll
    for (int t = 0; t < 4; ++t) {
        const int n0 = wave * 64 + t * 16;
        v8f acc = {};
        for (int k0 = 0; k0 < 64; k0 += 4) {
            int ka = k0 + half * 2;
            v2f a, bf;
            a[0] = sh[l15][ka];  a[1] = sh[l15][ka + 1];
            const float* wp = w2 + (n0 + l15) * 64 + ka;
            bf[0] = wp[0];       bf[1] = wp[1];
            acc = __builtin_amdgcn_wmma_f32_16x16x4_f32(
                    false, a, false, bf, (short)0, acc, false, false);
        }
        float bias = b2[n0 + l15];
        #pragma unroll
        for (int r = 0; r < 8; ++r) {
            int row = r + 8 * half;
            sx[row][n0 + l15] = acc[r] + bias;
        }
    }
    __syncthreads();

    // ---- per-row L2 norm: x/max(n,1e-12); uniform iff n < 1e-20
    {
        int row = tid >> 3, seg = tid & 7;
        float ss = 0.f;
        #pragma unroll 8
        for (int j = 0; j < 32; ++j) { float v = sx[row][seg * 32 + j]; ss += v * v; }
        srow[row][seg] = ss;
    }
    __syncthreads();
    if (tid < 16) {
        float ss = 0.f;
        #pragma unroll
        for (int j = 0; j < 8; ++j) ss += srow[tid][j];
        float n = sqrtf(ss);
        suni[tid]   = (n < 1e-20f) ? 1 : 0;
        sscale[tid] = 1.f / fmaxf(n, 1e-12f);
    }
    __syncthreads();
    for (int idx = tid; idx < 16 * SD; idx += 128) {
        int row = idx >> 8, c = idx & 255;
        float v = suni[row] ? 0.0625f : sx[row][c] * sscale[row];
        xout[(m0 + row) * SD + c] = v;
    }
}

// ---------------------------------------------------------------------------
// Kernel 3: 8-qubit / 3-layer state-vector circuit + <Z_q> + MLP head.
// One wave per batch row (8 rows per 256-thread block), psi in LDS.
// ---------------------------------------------------------------------------
__global__ __launch_bounds__(256) void circuit_head_kernel(
    const float* __restrict__ xn,  const float* __restrict__ qw,
    const float* __restrict__ scal,
    const float* __restrict__ hw1, const float* __restrict__ hb1,
    const float* __restrict__ hg,  const float* __restrict__ hbb,
    const float* __restrict__ hw2, const float* __restrict__ hb2,
    float* __restrict__ out)
{
    __shared__ float re[8][256];
    __shared__ float im[8][256];
    const int tid  = threadIdx.x;
    const int lane = tid & 31, wave = tid >> 5;
    const int row  = blockIdx.x * 8 + wave;

    const float* src = xn + row * SD;
    for (int j = lane; j < SD; j += 32) { re[wave][j] = src[j]; im[wave][j] = 0.f; }
    __syncthreads();

    for (int l = 0; l < 3; ++l) {
        for (int q = 0; q < 8; ++q) {
            const float* g = qw + (l * 8 + q) * 3;
            float phi = g[0], th = g[1], om = g[2];
            float ch = cosf(0.5f * th), shh = sinf(0.5f * th);
            float aa = 0.5f * (phi + om), bb = 0.5f * (phi - om);
            float ca = cosf(aa), sa = sinf(aa), cb = cosf(bb), sb = sinf(bb);
            float u00r =  ca * ch,  u00i = -sa * ch;
            float u01r = -cb * shh, u01i = -sb * shh;
            float u10r =  cb * shh, u10i =  sb * shh;
            float u11r =  ca * ch,  u11i =  sa * ch;
            int bit = 7 - q;
            #pragma unroll
            for (int jj = 0; jj < 4; ++jj) {
                int p  = lane + jj * 32;
                int i0 = ((p >> bit) << (bit + 1)) | (p & ((1 << bit) - 1));
                int i1 = i0 | (1 << bit);
                float ar = re[wave][i0], ai = im[wave][i0];
                float br = re[wave][i1], bi = im[wave][i1];
                re[wave][i0] = u00r*ar - u00i*ai + u01r*br - u01i*bi;
                im[wave][i0] = u00r*ai + u00i*ar + u01r*bi + u01i*br;
                re[wave][i1] = u10r*ar - u10i*ai + u11r*br - u11i*bi;
                im[wave][i1] = u10r*ai + u10i*ar + u11r*bi + u11i*br;
            }
            __syncthreads();
        }
        for (int c = 0; c < 8; ++c) {
            int t  = (c + 1) & 7;
            int bc = 7 - c, bt = 7 - t;
            int hi = bc > bt ? bc : bt, lo = bc > bt ? bt : bc;
            #pragma unroll
            for (int jj = 0; jj < 2; ++jj) {
                int p = lane + jj * 32;
                int i = ((p >> lo) << (lo + 1)) | (p & ((1 << lo) - 1));
                i = ((i >> hi) << (hi + 1)) | (i & ((1 << hi) - 1));
                i |= (1 << bc);
                int j2 = i | (1 << bt);
                float tr = re[wave][i], ti = im[wave][i];
                re[wave][i]  = re[wave][j2]; im[wave][i]  = im[wave][j2];
                re[wave][j2] = tr;           im[wave][j2] = ti;
            }
            __syncthreads();
        }
    }

    float z[8] = {0.f,0.f,0.f,0.f,0.f,0.f,0.f,0.f};
    #pragma unroll
    for (int jj = 0; jj < 8; ++jj) {
        int i = lane * 8 + jj;
        float pr = re[wave][i]*re[wave][i] + im[wave][i]*im[wave][i];
        #pragma unroll
        for (int q = 0; q < 8; ++q)
            z[q] += ((i >> (7 - q)) & 1) ? -pr : pr;
    }
    #pragma unroll
    for (int q = 0; q < 8; ++q)
        for (int off = 16; off > 0; off >>= 1)
            z[q] += __shfl_xor(z[q], off, 32);

    float v[14];
    #pragma unroll
    for (int q = 0; q < 8; ++q) v[q] = z[q];
    #pragma unroll
    for (int s = 0; s < 6; ++s) v[8 + s] = scal[row * 6 + s];
    float acc = hb1[lane];
    #pragma unroll
    for (int k = 0; k < 14; ++k) acc = fmaf(hw1[lane * 14 + k], v[k], acc);
    const float bninv = rsqrtf(1.f + 1e-5f);
    float h = fmaf(hg[lane] * bninv, acc, hbb[lane]);
    h = h > 0.f ? h : 0.f;
    #pragma unroll
    for (int o = 0; o < 3; ++o) {
        float part = hw2[o * 32 + lane] * h;
        for (int off = 16; off > 0; off >>= 1)
            part += __shfl_xor(part, off, 32);
        if (lane == 0) out[row * 3 + o] = part + hb2[o];
    }
}

// ---------------------------------------------------------------------------
extern "C" void kernel_launch(void* const* d_in, const int* in_sizes, int n_in,
                              void* d_out, int out_size, void* d_ws, size_t ws_size,
                              hipStream_t stream)
{
    const float* flux      = (const float*)d_in[0];
    const float* scalars   = (const float*)d_in[1];
    const float* conv1_w   = (const float*)d_in[2];
    const float* bn1_g     = (const float*)d_in[3];
    const float* bn1_b     = (const float*)d_in[4];
    const float* conv2_w   = (const float*)d_in[5];
    const float* bn2_g     = (const float*)d_in[6];
    const float* bn2_b     = (const float*)d_in[7];
    const float* proj_w1   = (const float*)d_in[8];
    const float* proj_b1   = (const float*)d_in[9];
    const float* proj_w2   = (const float*)d_in[10];
    const float* proj_b2   = (const float*)d_in[11];
    const float* q_weights = (const float*)d_in[12];
    const float* head_w1   = (const float*)d_in[13];
    const float* head_b1   = (const float*)d_in[14];
    const float* head_bn_g = (const float*)d_in[15];
    const float* head_bn_b = (const float*)d_in[16];
    const float* head_w2   = (const float*)d_in[17];
    const float* head_b2   = (const float*)d_in[18];
    float* outp  = (float*)d_out;
    float* feat  = (float*)d_ws;                 // 4096*256 floats
    float* xnorm = feat + (size_t)B_TOT * SD;    // 4096*256 floats

    extract_kernel<<<B_TOT, 256, 0, stream>>>(flux, conv1_w, bn1_g, bn1_b,
                                              conv2_w, bn2_g, bn2_b, feat);
    proj_norm_kernel<<<B_TOT / 16, 128, 0, stream>>>(feat, proj_w1, proj_b1,
                                                     proj_w2, proj_b2, xnorm);
    circuit_head_kernel<<<B_TOT / 8, 256, 0, stream>>>(xnorm, q_weights, scalars,
                                                       head_w1, head_b1,
                                                       head_bn_g, head_bn_b,
                                                       head_w2, head_b2, outp);
}